// GAT_61048665145866
// MI455X (gfx1250) — compile-verified
//
#include <hip/hip_runtime.h>
#include <math.h>

// ---------------- problem constants (match reference) ----------------
#define N_NODES   20000
#define N_EDGES   320000
#define E_TOT     (N_EDGES + N_NODES)   // edges + self loops = 340000
#define N_FEAT    128
#define HID       64
#define HEADS     4
#define F1_OUT    (HEADS * HID)         // 256
#define N_CLS     16
#define NEG_SLOPE 0.2f

typedef __attribute__((ext_vector_type(2))) float v2f;
typedef __attribute__((ext_vector_type(8))) float v8f;

// ---------------- helpers ----------------
__device__ __forceinline__ void atomicMaxF32(float* addr, float v) {
    // ordered-int trick: works with init = -inf, mixed sign updates
    if (v >= 0.0f) {
        atomicMax((int*)addr, __float_as_int(v));
    } else {
        atomicMin((unsigned int*)addr, __float_as_uint(v));
    }
}

__global__ void k_fill(float* p, float v, int n) {
    int i = blockIdx.x * blockDim.x + threadIdx.x;
    if (i < n) p[i] = v;
}

// out[n*C + c] = bias[c]
__global__ void k_bias_bcast(float* out, const float* bias, int C, int total) {
    int i = blockIdx.x * blockDim.x + threadIdx.x;
    if (i < total) out[i] = bias[i % C];
}

// ---------------- WMMA f32 GEMM: C[M,N] = A[M,K] @ B[K,N] ----------------
// Compile-time K (KC) and N (NC). One wave computes MT row sub-tiles of one
// 16-wide column tile (MT*16 x 16 output), reusing the B fragment MT times
// per k-step. MT must divide M/16 (no tail -> no divergent guards around
// WMMA, EXEC stays all-1s). V_WMMA_F32_16X16X4_F32, K stepped by 4.
template <int KC, int NC, int MT>
__global__ void k_gemm_wmma(const float* __restrict__ A,
                            const float* __restrict__ B,
                            float* __restrict__ C, int M) {
    const int lane    = threadIdx.x & 31;
    const int waveId  = blockIdx.x * (blockDim.x >> 5) + (threadIdx.x >> 5);
    const int tilesN  = NC / 16;
    const int groupsM = (M / 16) / MT;             // exact division by design
    if (waveId >= groupsM * tilesN) return;
    const int gm = waveId / tilesN;
    const int tn = waveId % tilesN;

    const int p = lane >> 4;   // half-wave: K sub-pair / output row group
    const int q = lane & 15;   // 0..15

    const int row0 = gm * MT * 16;

    // A: lane's base row (row0 + q); sub-tile i adds i*16 rows (const offset).
    const float* ab = A + (size_t)(row0 + q) * KC;
    // B: lane's column (tn*16 + q), rows k0+2p, k0+2p+1 (strided NC).
    const float* bc = B + tn * 16 + q;

    v8f acc[MT];
#pragma unroll
    for (int i = 0; i < MT; ++i) acc[i] = (v8f){};

#pragma unroll 4
    for (int k0 = 0; k0 < KC; k0 += 4) {
        // speculative prefetch of the A row stream ahead (global_prefetch_b8);
        // OOB speculative prefetches are silently dropped by hardware.
        __builtin_prefetch(ab + k0 + 64, 0, 3);

        v2f b;
        b.x = bc[(size_t)(k0 + 2 * p) * NC];
        b.y = bc[(size_t)(k0 + 2 * p + 1) * NC];

#pragma unroll
        for (int i = 0; i < MT; ++i) {
            v2f a = *(const v2f*)(ab + (size_t)i * 16 * KC + k0 + 2 * p);
            acc[i] = __builtin_amdgcn_wmma_f32_16x16x4_f32(
                /*neg_a=*/false, a, /*neg_b=*/false, b,
                /*c_mod=*/(short)0, acc[i],
                /*reuse_a=*/false, /*reuse_b=*/false);
        }
    }

    // D layout: VGPR r -> row (sub-tile base + r + 8*p), col tn*16 + q
#pragma unroll
    for (int i = 0; i < MT; ++i) {
        float* crow = C + (size_t)(row0 + i * 16 + 8 * p) * NC + tn * 16 + q;
#pragma unroll
        for (int r = 0; r < 8; ++r) crow[(size_t)r * NC] = acc[i][r];
    }
}

// ---------------- attention logits, layer 1: wave per (node, head) ----------------
// al_s[n,h] = sum_c h1[n, h*64+c] * a_src[h, c]   (same for dst)
__global__ void k_alpha1(const float* __restrict__ h1,
                         const float* __restrict__ a_src,
                         const float* __restrict__ a_dst,
                         float* __restrict__ al_s, float* __restrict__ al_d) {
    const int lane   = threadIdx.x & 31;
    const int waveId = blockIdx.x * (blockDim.x >> 5) + (threadIdx.x >> 5);
    if (waveId >= N_NODES * HEADS) return;
    const int n = waveId / HEADS;
    const int h = waveId % HEADS;

    const float* hp = h1 + (size_t)n * F1_OUT + h * HID;
    const float* as = a_src + h * HID;
    const float* ad = a_dst + h * HID;

    float v0 = hp[lane], v1 = hp[lane + 32];
    float vs = v0 * as[lane] + v1 * as[lane + 32];
    float vd = v0 * ad[lane] + v1 * ad[lane + 32];
#pragma unroll
    for (int off = 16; off > 0; off >>= 1) {
        vs += __shfl_down(vs, off, 32);
        vd += __shfl_down(vd, off, 32);
    }
    if (lane == 0) {
        al_s[n * HEADS + h] = vs;
        al_d[n * HEADS + h] = vd;
    }
}

// ---------------- attention logits, layer 2: thread per node ----------------
__global__ void k_alpha2(const float* __restrict__ t2,
                         const float* __restrict__ a_src,
                         const float* __restrict__ a_dst,
                         float* __restrict__ al_s, float* __restrict__ al_d) {
    int n = blockIdx.x * blockDim.x + threadIdx.x;
    if (n >= N_NODES) return;
    const float* tp = t2 + (size_t)n * N_CLS;
    float vs = 0.f, vd = 0.f;
#pragma unroll
    for (int c = 0; c < N_CLS; ++c) {
        float t = tp[c];
        vs += t * a_src[c];
        vd += t * a_dst[c];
    }
    al_s[n] = vs;
    al_d[n] = vd;
}

// ---------------- edge helpers ----------------
__device__ __forceinline__ void edge_nodes(const long long* ei, int e, int& src, int& dst) {
    if (e < N_EDGES) {
        src = (int)ei[e];
        dst = (int)ei[N_EDGES + e];
    } else {
        src = dst = e - N_EDGES;   // self loop
    }
}

// pass 1: segment max of leaky-relu logits. thread per (edge, head)
__global__ void k_edge_max(const long long* __restrict__ ei,
                           const float* __restrict__ al_s,
                           const float* __restrict__ al_d,
                           float* __restrict__ m, int H) {
    int t = blockIdx.x * blockDim.x + threadIdx.x;
    if (t >= E_TOT * H) return;
    int e = t / H, h = t % H;
    int src, dst; edge_nodes(ei, e, src, dst);
    float le = al_s[src * H + h] + al_d[dst * H + h];
    le = (le > 0.f) ? le : NEG_SLOPE * le;
    atomicMaxF32(&m[dst * H + h], le);
}

// pass 2: exp(le - max) stored per edge, segment sum. thread per (edge, head)
__global__ void k_edge_exp(const long long* __restrict__ ei,
                           const float* __restrict__ al_s,
                           const float* __restrict__ al_d,
                           const float* __restrict__ m,
                           float* __restrict__ pexp,
                           float* __restrict__ s, int H) {
    int t = blockIdx.x * blockDim.x + threadIdx.x;
    if (t >= E_TOT * H) return;
    int e = t / H, h = t % H;
    int src, dst; edge_nodes(ei, e, src, dst);
    float le = al_s[src * H + h] + al_d[dst * H + h];
    le = (le > 0.f) ? le : NEG_SLOPE * le;
    float pe = expf(le - m[dst * H + h]);
    pexp[e * H + h] = pe;
    atomicAdd(&s[dst * H + h], pe);
}

// pass 3: weighted scatter. thread per (edge, channel); loops heads.
__global__ void k_edge_agg(const long long* __restrict__ ei,
                           const float* __restrict__ hfeat,   // [N, H*C]
                           const float* __restrict__ pexp,    // [E_TOT, H]
                           const float* __restrict__ s,       // [N, H]
                           float* __restrict__ agg,           // [N, H*C]
                           int H, int C) {
    int t = blockIdx.x * blockDim.x + threadIdx.x;
    if (t >= E_TOT * C) return;
    int e = t / C, c = t % C;
    int src, dst; edge_nodes(ei, e, src, dst);
    for (int h = 0; h < H; ++h) {
        float alpha = pexp[e * H + h] / (s[dst * H + h] + 1e-16f);
        float v = hfeat[(size_t)src * (H * C) + h * C + c] * alpha;
        atomicAdd(&agg[(size_t)dst * (H * C) + h * C + c], v);
    }
}

// h = elu(agg + b1)   over N*256
__global__ void k_bias_elu(const float* __restrict__ agg,
                           const float* __restrict__ b1,
                           float* __restrict__ out) {
    int i = blockIdx.x * blockDim.x + threadIdx.x;
    if (i >= N_NODES * F1_OUT) return;
    float v = agg[i] + b1[i % F1_OUT];
    out[i] = (v > 0.f) ? v : (expf(v) - 1.0f);
}

// ---------------- launcher ----------------
extern "C" void kernel_launch(void* const* d_in, const int* in_sizes, int n_in,
                              void* d_out, int out_size, void* d_ws, size_t ws_size,
                              hipStream_t stream) {
    const float*     x      = (const float*)d_in[0];
    const long long* ei     = (const long long*)d_in[1];   // int64 [2, E]
    const float*     W1     = (const float*)d_in[2];
    const float*     a_src1 = (const float*)d_in[3];
    const float*     a_dst1 = (const float*)d_in[4];
    const float*     b1     = (const float*)d_in[5];
    const float*     W2     = (const float*)d_in[6];
    const float*     a_src2 = (const float*)d_in[7];
    const float*     a_dst2 = (const float*)d_in[8];
    const float*     b2     = (const float*)d_in[9];
    float* out = (float*)d_out;

    // workspace layout (floats)
    float* w     = (float*)d_ws;
    float* h1    = w;                           // N*256 (later reused for ELU output)
    float* agg1  = h1    + (size_t)N_NODES * F1_OUT;   // N*256
    float* t2    = agg1  + (size_t)N_NODES * F1_OUT;   // N*16
    float* al_s1 = t2    + (size_t)N_NODES * N_CLS;    // N*4
    float* al_d1 = al_s1 + (size_t)N_NODES * HEADS;
    float* m1    = al_d1 + (size_t)N_NODES * HEADS;
    float* s1    = m1    + (size_t)N_NODES * HEADS;
    float* pexp1 = s1    + (size_t)N_NODES * HEADS;    // E_TOT*4
    float* al_s2 = pexp1 + (size_t)E_TOT * HEADS;      // N
    float* al_d2 = al_s2 + N_NODES;
    float* m2    = al_d2 + N_NODES;
    float* s2    = m2    + N_NODES;
    float* pexp2 = s2    + N_NODES;                    // E_TOT

    const int B = 256;
    auto blocks = [](long long n, int b) { return (int)((n + b - 1) / b); };

    const int MT      = 5;                       // 1250 row tiles = 250 groups, no tail
    const int groupsM = (N_NODES / 16) / MT;     // 250

    // ---- layer 1 ----
    k_fill<<<blocks(N_NODES * HEADS, B), B, 0, stream>>>(m1, -INFINITY, N_NODES * HEADS);
    k_fill<<<blocks(N_NODES * HEADS, B), B, 0, stream>>>(s1, 0.f, N_NODES * HEADS);
    k_fill<<<blocks((long long)N_NODES * F1_OUT, B), B, 0, stream>>>(agg1, 0.f, N_NODES * F1_OUT);

    {   // h1 = x @ W1  : waves = groupsM * (256/16)
        long long waves = (long long)groupsM * (F1_OUT / 16);
        k_gemm_wmma<N_FEAT, F1_OUT, 5>
            <<<blocks(waves * 32, B), B, 0, stream>>>(x, W1, h1, N_NODES);
    }
    {   // logits: wave per (node, head)
        long long waves = (long long)N_NODES * HEADS;
        k_alpha1<<<blocks(waves * 32, B), B, 0, stream>>>(h1, a_src1, a_dst1, al_s1, al_d1);
    }
    k_edge_max<<<blocks((long long)E_TOT * HEADS, B), B, 0, stream>>>(ei, al_s1, al_d1, m1, HEADS);
    k_edge_exp<<<blocks((long long)E_TOT * HEADS, B), B, 0, stream>>>(ei, al_s1, al_d1, m1, pexp1, s1, HEADS);
    k_edge_agg<<<blocks((long long)E_TOT * HID, B), B, 0, stream>>>(ei, h1, pexp1, s1, agg1, HEADS, HID);

    // h = elu(agg1 + b1), written into h1's buffer (h1 no longer needed)
    k_bias_elu<<<blocks((long long)N_NODES * F1_OUT, B), B, 0, stream>>>(agg1, b1, h1);

    // ---- layer 2 ----
    k_fill<<<blocks(N_NODES, B), B, 0, stream>>>(m2, -INFINITY, N_NODES);
    k_fill<<<blocks(N_NODES, B), B, 0, stream>>>(s2, 0.f, N_NODES);
    k_bias_bcast<<<blocks((long long)N_NODES * N_CLS, B), B, 0, stream>>>(out, b2, N_CLS, N_NODES * N_CLS);

    {   // t2 = h @ W2 : waves = groupsM * 1
        long long waves = (long long)groupsM;
        k_gemm_wmma<F1_OUT, N_CLS, 5>
            <<<blocks(waves * 32, B), B, 0, stream>>>(h1, W2, t2, N_NODES);
    }
    k_alpha2<<<blocks(N_NODES, B), B, 0, stream>>>(t2, a_src2, a_dst2, al_s2, al_d2);

    k_edge_max<<<blocks((long long)E_TOT, B), B, 0, stream>>>(ei, al_s2, al_d2, m2, 1);
    k_edge_exp<<<blocks((long long)E_TOT, B), B, 0, stream>>>(ei, al_s2, al_d2, m2, pexp2, s2, 1);
    // scatter directly into bias-initialized d_out
    k_edge_agg<<<blocks((long long)E_TOT * N_CLS, B), B, 0, stream>>>(ei, t2, pexp2, s2, out, 1, N_CLS);
}